// EdgeConv_5549097746955
// MI455X (gfx1250) — compile-verified
//
#include <hip/hip_runtime.h>
#include <hip/hip_bf16.h>

// EdgeConv (N=120000, K=16, F=64) refactored:
//   T = feat @ theta_w                      [N,64]
//   E = feat @ theta_w + feat @ phi_w + b   [N,64]
//   out[v,f] = E[v,f] - min_k T[nbr[v,k], f]
// Kernel 1: fp32 WMMA GEMM (v_wmma_f32_16x16x4_f32), 16 rows/wave, 64 cols.
// Kernel 2: wave-per-node gather-min over K=16 neighbors (L2-resident T).

typedef __attribute__((ext_vector_type(2))) float v2f;
typedef __attribute__((ext_vector_type(8))) float v8f;

#define N_NODES 120000
#define KNBR    16
#define FDIM    64

__global__ __launch_bounds__(128) void edgeconv_gemm_wmma(
    const float* __restrict__ feat,     // [N,64]
    const float* __restrict__ theta_w,  // [64,64] row-major (fin, fout)
    const float* __restrict__ theta_b,  // [64]
    const float* __restrict__ phi_w,    // [64,64]
    const float* __restrict__ phi_b,    // [64]
    float* __restrict__ T,              // [N,64] out
    float* __restrict__ E)              // [N,64] out
{
    const int lane    = threadIdx.x & 31;
    const int wave    = threadIdx.x >> 5;
    const int rowBase = (blockIdx.x * 4 + wave) * 16;   // 16 nodes per wave
    const int half    = lane >> 4;                      // 0: lanes 0-15, 1: lanes 16-31
    const int l       = lane & 15;

    v8f accT[4] = {};   // feat @ theta_w, 4 feature tiles of 16
    v8f accP[4] = {};   // feat @ phi_w

    // K = 64 in steps of 4 (WMMA f32 16x16x4)
    #pragma unroll
    for (int kk = 0; kk < 16; ++kk) {
        const int k0 = kk * 4 + half * 2;
        // A fragment 16x4: lane half selects K pair, l selects M row
        const float* arow = feat + (size_t)(rowBase + l) * FDIM + k0;
        v2f a;
        a.x = arow[0];
        a.y = arow[1];
        #pragma unroll
        for (int ft = 0; ft < 4; ++ft) {
            // B fragment 4x16: row k0 / k0+1, col = ft*16 + l
            const float* bt = theta_w + (size_t)k0 * FDIM + ft * 16 + l;
            const float* bp = phi_w   + (size_t)k0 * FDIM + ft * 16 + l;
            v2f wb, pb;
            wb.x = bt[0]; wb.y = bt[FDIM];
            pb.x = bp[0]; pb.y = bp[FDIM];
            accT[ft] = __builtin_amdgcn_wmma_f32_16x16x4_f32(
                false, a, false, wb, (short)0, accT[ft], false, false);
            accP[ft] = __builtin_amdgcn_wmma_f32_16x16x4_f32(
                false, a, false, pb, (short)0, accP[ft], false, false);
        }
    }

    // C/D layout: VGPR r -> M = r (lanes 0-15) or r+8 (lanes 16-31), N = l
    #pragma unroll
    for (int ft = 0; ft < 4; ++ft) {
        const int col  = ft * 16 + l;
        const float bias = theta_b[col] + phi_b[col];
        #pragma unroll
        for (int r = 0; r < 8; ++r) {
            const int row = rowBase + r + half * 8;
            const float t = accT[ft][r];
            T[(size_t)row * FDIM + col] = t;
            E[(size_t)row * FDIM + col] = t + accP[ft][r] + bias;
        }
    }
}

__global__ __launch_bounds__(256) void edgeconv_gather_min(
    const long long* __restrict__ nbr,  // [N,16] int64
    const float* __restrict__ T,        // [N,64] (L2-resident)
    const float* __restrict__ E,        // [N,64]
    float* __restrict__ out)            // [N,64]
{
    const int lane = threadIdx.x & 31;
    const int wave = threadIdx.x >> 5;
    const int v    = blockIdx.x * 8 + wave;   // one node per wave

    const long long* idx = nbr + (size_t)v * KNBR;

    float2 m = make_float2(3.402823466e+38f, 3.402823466e+38f);

    long long u = idx[0];
    #pragma unroll
    for (int k = 0; k < KNBR; ++k) {
        const long long unext = (k + 1 < KNBR) ? idx[k + 1] : u;
        // prefetch next neighbor's row while reducing this one
        __builtin_prefetch(T + (size_t)unext * FDIM + lane * 2, 0, 0);
        const float2 t = *(const float2*)(T + (size_t)u * FDIM + lane * 2);
        m.x = fminf(m.x, t.x);
        m.y = fminf(m.y, t.y);
        u = unext;
    }

    const float2 e = *(const float2*)(E + (size_t)v * FDIM + lane * 2);
    float2 o;
    o.x = e.x - m.x;
    o.y = e.y - m.y;
    *(float2*)(out + (size_t)v * FDIM + lane * 2) = o;
}

extern "C" void kernel_launch(void* const* d_in, const int* in_sizes, int n_in,
                              void* d_out, int out_size, void* d_ws, size_t ws_size,
                              hipStream_t stream) {
    const float*     feat    = (const float*)d_in[0];
    const long long* nbr_idx = (const long long*)d_in[1];
    const float*     theta_w = (const float*)d_in[2];
    const float*     theta_b = (const float*)d_in[3];
    const float*     phi_w   = (const float*)d_in[4];
    const float*     phi_b   = (const float*)d_in[5];
    float*           out     = (float*)d_out;

    float* T = (float*)d_ws;                                   // N*64 floats
    float* E = (float*)d_ws + (size_t)N_NODES * FDIM;          // N*64 floats

    // GEMM: 16 nodes per wave, 4 waves (128 threads) per block -> 64 nodes/block
    // 120000 / 64 = 1875 blocks (exact)
    edgeconv_gemm_wmma<<<1875, 128, 0, stream>>>(
        feat, theta_w, theta_b, phi_w, phi_b, T, E);

    // Gather-min: 1 node per wave, 8 waves/block -> 120000 / 8 = 15000 blocks
    edgeconv_gather_min<<<15000, 256, 0, stream>>>(nbr_idx, T, E, out);
}